// STATS5_38774964748816
// MI455X (gfx1250) — compile-verified
//
#include <hip/hip_runtime.h>
#include <math.h>

// ---------------------------------------------------------------------------
// Batched 12-state RK4 ODE (Epo receptor model) for MI455X / gfx1250.
// One wave32 lane per batch element; everything in VGPRs; forcing table is a
// compile-time __constant__ (uniform index -> scalar SMEM loads); observation
// interpolation fused into the time loop; true+noised outputs fused.
// ---------------------------------------------------------------------------

#define NSTEP 69           // N_T - 1 integration steps
#define OBS   8            // design points
#define SDIM  12           // state dimension

// ---- compile-time E(t) forcing table (jnp.interp of fixed 16-pt table) ----
struct ETab { float lo[NSTEP]; float mid[NSTEP]; float hi[NSTEP]; };

constexpr double EP_X[16] = {0,2,4,6,8,10,12,14,16,18,20,25,30,40,50,60};
constexpr double EP_Y[16] = {0.01713,0.145,0.2442,0.7659,1.0,0.8605,0.7829,0.5705,
                             0.6217,0.331,0.3388,0.3116,0.05062,0.02504,0.01163,0.01163};

constexpr float epo_interp(double t) {
  if (t <= EP_X[0])  return (float)EP_Y[0];
  if (t >= EP_X[15]) return (float)EP_Y[15];
  for (int i = 0; i < 15; ++i) {
    if (t <= EP_X[i + 1]) {
      double w = (t - EP_X[i]) / (EP_X[i + 1] - EP_X[i]);
      return (float)(EP_Y[i] + (EP_Y[i + 1] - EP_Y[i]) * w);
    }
  }
  return (float)EP_Y[15];
}

constexpr ETab make_etab() {
  ETab e{};
  constexpr double dt = 60.0 / 69.0;
  for (int i = 0; i < NSTEP; ++i) {
    double t = dt * (double)i;
    e.lo[i]  = epo_interp(t);
    e.mid[i] = epo_interp(t + 0.5 * dt);
    e.hi[i]  = epo_interp(t + dt);
  }
  return e;
}

static __constant__ ETab E_TAB = make_etab();

// ---- ODE right-hand side (12 states, all in registers) ----
__device__ __forceinline__ void deriv(const float* __restrict__ x,
                                      float k1, float k2, float r, float E,
                                      float* __restrict__ d) {
  const float x1sq  = x[1] * x[1];
  const float k1x0E = k1 * x[0] * E;
  d[0] = -k1x0E + k2 * x[11];
  d[1] = -x1sq + k1x0E;
  d[2] = -k2 * x[2] + x1sq;
  d[3] = k2 * (x[2] - x[11]);
  d[4] = r * (x[2] - x[4]);
#pragma unroll
  for (int i = 5; i < SDIM; ++i) d[i] = r * (x[i - 1] - x[i]);
}

__global__ __launch_bounds__(256) void epo_rk4_kernel(
    const float* __restrict__ params,   // (B,3)
    const float* __restrict__ design,   // (8,)
    const float* __restrict__ noise,    // (B,16)
    float* __restrict__ out,            // (B,16) true then (B,16) noised
    int B) {
  const int b = blockIdx.x * blockDim.x + threadIdx.x;

  // stage the 8 shared design points in LDS (emits gfx1250 split-barrier)
  __shared__ float s_des[OBS];
  if (threadIdx.x < OBS) s_des[threadIdx.x] = design[threadIdx.x];
  __syncthreads();

  if (b >= B) return;

  const size_t ob = (size_t)b * 16;
  // pull noise lines in under the compute shadow (global_prefetch_b8)
  __builtin_prefetch(&noise[ob], 0, 3);

  // ---- parameter transform: shift + normcdf(p)*scale ----
  const float INV_SQRT2 = 0.7071067811865476f;
  const float p0 = params[(size_t)b * 3 + 0];
  const float p1 = params[(size_t)b * 3 + 1];
  const float p2 = params[(size_t)b * 3 + 2];
  const float k1  = 0.5f  + 2.5f  * (0.5f * erfcf(-p0 * INV_SQRT2));
  const float k2  = 0.05f + 0.15f * (0.5f * erfcf(-p1 * INV_SQRT2));
  const float tau = 4.0f  + 6.0f  * (0.5f * erfcf(-p2 * INV_SQRT2));
  const float r   = 8.0f / tau;

  const float dt     = 60.0f / 69.0f;
  const float inv_dt = 69.0f / 60.0f;
  const float hdt    = 0.5f * dt;
  const float dt6    = dt / 6.0f;

  // ---- design-point interp indices/weights on the uniform T0 grid ----
  int   jj[OBS];
  float ww[OBS];
#pragma unroll
  for (int k = 0; k < OBS; ++k) {
    float pos = s_des[k] * inv_dt;
    int   j   = (int)floorf(pos);
    if (j < 0) j = 0;
    if (j > NSTEP - 1) j = NSTEP - 1;
    float w = pos - (float)j;
    if (w < 0.0f) w = 0.0f;
    if (w > 1.0f) w = 1.0f;
    jj[k] = j;
    ww[k] = w;
  }

  // ---- state + fused observation accumulators ----
  float x[SDIM];
#pragma unroll
  for (int i = 0; i < SDIM; ++i) x[i] = 0.0f;
  x[0] = 3.71f;

  float a1[OBS], a2[OBS];
#pragma unroll
  for (int k = 0; k < OBS; ++k) { a1[k] = 0.0f; a2[k] = 0.0f; }

  // contribution of time index 0 (initial state)
  {
    const float y1 = 0.0f;                 // 0.33*(x1+x2) = 0 at t=0
    const float y2 = 0.26f * 3.71f;
#pragma unroll
    for (int k = 0; k < OBS; ++k) {
      float c = (jj[k] == 0 ? (1.0f - ww[k]) : 0.0f);
      a1[k] = fmaf(c, y1, a1[k]);
      a2[k] = fmaf(c, y2, a2[k]);
    }
  }

  // ---- RK4 time loop ----
  float s1[SDIM], s2[SDIM], s3[SDIM], s4[SDIM], xt[SDIM];
  for (int n = 0; n < NSTEP; ++n) {
    const float ea = E_TAB.lo[n];          // uniform index -> s_load
    const float eb = E_TAB.mid[n];
    const float ec = E_TAB.hi[n];

    deriv(x, k1, k2, r, ea, s1);
#pragma unroll
    for (int i = 0; i < SDIM; ++i) xt[i] = fmaf(hdt, s1[i], x[i]);
    deriv(xt, k1, k2, r, eb, s2);
#pragma unroll
    for (int i = 0; i < SDIM; ++i) xt[i] = fmaf(hdt, s2[i], x[i]);
    deriv(xt, k1, k2, r, eb, s3);
#pragma unroll
    for (int i = 0; i < SDIM; ++i) xt[i] = fmaf(dt, s3[i], x[i]);
    deriv(xt, k1, k2, r, ec, s4);
#pragma unroll
    for (int i = 0; i < SDIM; ++i)
      x[i] = fmaf(dt6, s1[i] + 2.0f * (s2[i] + s3[i]) + s4[i], x[i]);

    // fused observation interpolation at time index n+1
    const int   ti = n + 1;
    const float y1 = 0.33f * (x[1] + x[2]);
    const float y2 = 0.26f * (x[0] + x[1] + x[2]);
#pragma unroll
    for (int k = 0; k < OBS; ++k) {
      float c = (jj[k] == ti ? (1.0f - ww[k]) : 0.0f) +
                (jj[k] + 1 == ti ? ww[k] : 0.0f);
      a1[k] = fmaf(c, y1, a1[k]);
      a2[k] = fmaf(c, y2, a2[k]);
    }
  }

  // ---- fused output: true_obs then noised_obs ----
  float* __restrict__ tr = out;
  float* __restrict__ no = out + (size_t)B * 16;
#pragma unroll
  for (int k = 0; k < OBS; ++k) {
    const float t1 = a1[k];
    const float t2 = a2[k];
    tr[ob + k]     = t1;
    tr[ob + 8 + k] = t2;
    no[ob + k]     = fmaf(0.01f, noise[ob + k], t1);
    no[ob + 8 + k] = fmaf(0.01f, noise[ob + 8 + k], t2);
  }
}

extern "C" void kernel_launch(void* const* d_in, const int* in_sizes, int n_in,
                              void* d_out, int out_size, void* d_ws, size_t ws_size,
                              hipStream_t stream) {
  const float* params = (const float*)d_in[0];  // (B,3) fp32
  const float* design = (const float*)d_in[1];  // (8,)  fp32
  const float* noise  = (const float*)d_in[2];  // (B,16) fp32
  float*       out    = (float*)d_out;          // 2*(B,16) fp32

  const int B = in_sizes[0] / 3;
  const int threads = 256;                      // 8 wave32 waves / block
  const int blocks  = (B + threads - 1) / threads;
  epo_rk4_kernel<<<blocks, threads, 0, stream>>>(params, design, noise, out, B);
}